// LBMLayer_40415642255523
// MI455X (gfx1250) — compile-verified
//
#include <hip/hip_runtime.h>

// ---------------------------------------------------------------------------
// D2Q9 LBM, 1024x1024, 4 iterations — CDNA5 (gfx1250) WMMA implementation.
//
// Fusion: since M2W @ W2M == I (verified analytically), one collision step is
//   w_new = (1-1/tau) * w_streamed + (1/tau) * Minv @ eq
// which we fold into a single 16x20 x 20xN FP32 GEMM per 16-cell tile using
// V_WMMA_F32_16X16X4_F32 (5 K-steps of 4, K padded 18->20).
// ---------------------------------------------------------------------------

typedef float v2f __attribute__((ext_vector_type(2)));
typedef float v8f __attribute__((ext_vector_type(8)));

namespace {
constexpr int   NXg   = 1024;
constexpr int   NYg   = 1024;
constexpr int   TPB   = 256;        // 8 wave32 per block
constexpr int   BSTR  = 264;        // LDS row stride (floats); conflict-free pad
constexpr float TAUc    = 0.7f;
constexpr float TAU_INV = 1.0f / 0.7f;       //  10/7
constexpr float OMTI    = 1.0f - TAU_INV;    //  -3/7
constexpr float GXc = 0.0f;
constexpr float GYc = -0.0001f;
}

// Exact inverse of the reference moment matrix M (rows of M2W).
__constant__ float MINV[9][9] = {
  {4.0f/9.0f,       0.0f,       0.0f, -2.0f/3.0f, -2.0f/3.0f,  0.0f,  0.0f,  0.0f,  1.0f},
  {1.0f/9.0f,  1.0f/3.0f,       0.0f,  1.0f/3.0f, -1.0f/6.0f,  0.0f, -0.5f,  0.0f, -0.5f},
  {1.0f/9.0f,       0.0f,  1.0f/3.0f, -1.0f/6.0f,  1.0f/3.0f,  0.0f,  0.0f, -0.5f, -0.5f},
  {1.0f/9.0f, -1.0f/3.0f,       0.0f,  1.0f/3.0f, -1.0f/6.0f,  0.0f,  0.5f,  0.0f, -0.5f},
  {1.0f/9.0f,       0.0f, -1.0f/3.0f, -1.0f/6.0f,  1.0f/3.0f,  0.0f,  0.0f,  0.5f, -0.5f},
  {1.0f/36.0f,  1.0f/12.0f,  1.0f/12.0f, 1.0f/12.0f, 1.0f/12.0f,  0.25f,  0.25f,  0.25f, 0.25f},
  {1.0f/36.0f, -1.0f/12.0f,  1.0f/12.0f, 1.0f/12.0f, 1.0f/12.0f, -0.25f, -0.25f,  0.25f, 0.25f},
  {1.0f/36.0f, -1.0f/12.0f, -1.0f/12.0f, 1.0f/12.0f, 1.0f/12.0f,  0.25f, -0.25f, -0.25f, 0.25f},
  {1.0f/36.0f,  1.0f/12.0f, -1.0f/12.0f, 1.0f/12.0f, 1.0f/12.0f, -0.25f,  0.25f, -0.25f, 0.25f},
};

// Combined update matrix A (16 rows x 20 cols):
//   rows 0..8 :  [ OMTI * I9  |  TAU_INV * MINV  |  0 0 ]
//   rows 9..15:  0   (pad rows; their outputs are never stored)
__device__ __forceinline__ float acomb_val(int r, int k) {
  if (r >= 9 || k >= 18) return 0.0f;
  if (k < 9)  return (k == r) ? OMTI : 0.0f;
  return TAU_INV * MINV[r][k - 9];
}

__device__ __forceinline__ v8f wmma_f32_16x16x4(v2f a, v2f b, v8f c) {
#if __has_builtin(__builtin_amdgcn_wmma_f32_16x16x4_f32)
  // 8 args: (neg_a, A, neg_b, B, c_mod, C, reuse_a, reuse_b)
  return __builtin_amdgcn_wmma_f32_16x16x4_f32(false, a, false, b, (short)0, c,
                                               false, false);
#else
  v8f d;
  asm volatile("v_wmma_f32_16x16x4_f32 %0, %1, %2, %3"
               : "=v"(d)
               : "v"(a), "v"(b), "v"(c));
  return d;
#endif
}

__global__ __launch_bounds__(TPB)
void lbm_step(const float* __restrict__ src, float* __restrict__ dst,
              const float* __restrict__ wt) {
  __shared__ float Blds[20 * BSTR];  // WMMA B staging: [k=0..19][cell=0..255]
  __shared__ float Olds[TPB * 9];    // output staging, flat (cell*9 + comp)

  const int t = threadIdx.x;                 // 0..255  <-> local cell id
  const int g = blockIdx.x * TPB + t;        // global flat cell id
  const int x = g >> 10;
  const int y = g & (NYg - 1);

  // ---- streaming step: gather 9 populations with periodic wrap ----
  constexpr int EX[9] = {0, 1, 0, -1, 0, 1, -1, -1, 1};
  constexpr int EY[9] = {0, 0, 1, 0, -1, 1, 1, -1, -1};
  float s[9];
#pragma unroll
  for (int i = 0; i < 9; ++i) {
    const int xs = (x - EX[i]) & (NXg - 1);
    const int ys = (y - EY[i]) & (NYg - 1);
    s[i] = src[((xs << 10) + ys) * 9 + i];
  }

  // ---- per-cell macroscopic quantities + equilibrium moments ----
  const float rho = ((s[0] + s[1]) + (s[2] + s[3])) +
                    ((s[4] + s[5]) + (s[6] + s[7])) + s[8];
  const float mx = s[1] - s[3] + s[5] - s[6] - s[7] + s[8];
  const float my = s[2] - s[4] + s[5] + s[6] - s[7] - s[8];
  const float rinv = 1.0f / rho;
  const float ux = mx * rinv + TAUc * GXc * rho;
  const float uy = my * rinv + TAUc * GYc * rho;

  float eq[9];
  eq[0] = rho;
  eq[1] = rho * ux;
  eq[2] = rho * uy;
  eq[3] = wt[0] * rho * ux * ux;
  eq[4] = wt[1] * rho * uy * uy;
  eq[5] = wt[2] * rho * ux * uy;
  eq[6] = wt[3] * rho * ux * uy * uy;
  eq[7] = wt[4] * rho * ux * ux * uy;
  eq[8] = wt[5] * rho * ux * ux * uy * uy;

  // ---- stage B columns: [s0..s8, eq0..eq8, 0, 0] ----
#pragma unroll
  for (int k = 0; k < 9; ++k) Blds[k * BSTR + t] = s[k];
#pragma unroll
  for (int k = 0; k < 9; ++k) Blds[(9 + k) * BSTR + t] = eq[k];
  Blds[18 * BSTR + t] = 0.0f;
  Blds[19 * BSTR + t] = 0.0f;
  __syncthreads();

  // ---- WMMA phase: each wave updates its own 32 cells (2 tiles of 16) ----
  const int wave = t >> 5;
  const int lane = t & 31;
  const int col  = lane & 15;   // output column / A row index
  const int h    = lane >> 4;   // lane-half selects K pair (ISA 7.12.2)

  v2f a[5];
#pragma unroll
  for (int stp = 0; stp < 5; ++stp) {
    const int k = 4 * stp + 2 * h;
    a[stp].x = acomb_val(col, k);
    a[stp].y = acomb_val(col, k + 1);
  }

#pragma unroll
  for (int tile = 0; tile < 2; ++tile) {
    const int cb = wave * 32 + tile * 16;
    v8f c = {0.f, 0.f, 0.f, 0.f, 0.f, 0.f, 0.f, 0.f};
#pragma unroll
    for (int stp = 0; stp < 5; ++stp) {
      const int k0 = 4 * stp + 2 * h;
      v2f b;
      b.x = Blds[(k0    ) * BSTR + cb + col];
      b.y = Blds[(k0 + 1) * BSTR + cb + col];
      c = wmma_f32_16x16x4(a[stp], b, c);   // full EXEC here (uniform flow)
    }
    // D layout: VGPR v = row v (lanes 0-15) / row v+8 (lanes 16-31)
    const int cell = cb + col;
    if (h == 0) {
#pragma unroll
      for (int v = 0; v < 8; ++v) Olds[cell * 9 + v] = c[v];
    } else {
      Olds[cell * 9 + 8] = c[0];            // row 8; rows 9..15 are pad
    }
  }
  __syncthreads();

  // ---- fully coalesced writeback of this block's 2304 contiguous floats ----
  const size_t base = (size_t)blockIdx.x * (TPB * 9);
#pragma unroll
  for (int j = 0; j < 9; ++j) dst[base + j * TPB + t] = Olds[j * TPB + t];
}

// ---------------------------------------------------------------------------
// 4 iterations, ping-pong: d_in -> ws -> d_out -> ws -> d_out.
// Requires ws_size >= 1024*1024*9*4 = 37,748,736 bytes of scratch.
// ---------------------------------------------------------------------------
extern "C" void kernel_launch(void* const* d_in, const int* in_sizes, int n_in,
                              void* d_out, int out_size, void* d_ws,
                              size_t ws_size, hipStream_t stream) {
  (void)in_sizes; (void)n_in; (void)out_size; (void)ws_size;
  const float* in = (const float*)d_in[0];
  const float* wt = (const float*)d_in[1];
  float* out = (float*)d_out;
  float* ws  = (float*)d_ws;

  const dim3 grid((NXg * NYg) / TPB);
  const dim3 block(TPB);
  lbm_step<<<grid, block, 0, stream>>>(in,  ws,  wt);
  lbm_step<<<grid, block, 0, stream>>>(ws,  out, wt);
  lbm_step<<<grid, block, 0, stream>>>(out, ws,  wt);
  lbm_step<<<grid, block, 0, stream>>>(ws,  out, wt);
}